// SqueezeAndExcitation1D_1752346657530
// MI455X (gfx1250) — compile-verified
//
#include <hip/hip_runtime.h>
#include <math.h>

typedef __attribute__((ext_vector_type(2))) float v2f;
typedef __attribute__((ext_vector_type(8))) float v8f;

constexpr int B = 32;
constexpr int C = 512;   // IN_CHANNELS
constexpr int R = 256;   // REDUCED
constexpr int L = 4096;

// ---------------------------------------------------------------------------
// Kernel 1: squeeze — global average pool over L. One block per (b,c) row.
// 256 threads x 4 float4 loads = 4096 elements; wave32 shuffle reduction.
// ---------------------------------------------------------------------------
__global__ __launch_bounds__(256) void se_squeeze(const float* __restrict__ x,
                                                  float* __restrict__ S) {
    const int row = blockIdx.x;                       // b*C + c
    const float4* xr = (const float4*)(x + (size_t)row * L);
    const int tid = threadIdx.x;
    float sum = 0.f;
#pragma unroll
    for (int i = 0; i < 4; ++i) {
        float4 v = xr[i * 256 + tid];
        sum += (v.x + v.y) + (v.z + v.w);
    }
#pragma unroll
    for (int off = 16; off > 0; off >>= 1)
        sum += __shfl_down(sum, off, 32);
    __shared__ float wsum[8];
    const int wave = tid >> 5, lane = tid & 31;
    if (lane == 0) wsum[wave] = sum;
    __syncthreads();
    if (tid == 0) {
        float tot = 0.f;
#pragma unroll
        for (int i = 0; i < 8; ++i) tot += wsum[i];
        S[row] = tot * (1.0f / (float)L);
    }
}

// ---------------------------------------------------------------------------
// Kernel 2: excitation MLP on the matrix pipe (V_WMMA_F32_16X16X4_F32).
// One workgroup, 32 waves. GEMM1: H = relu(S[32,512] @ W1^T + b1) -> LDS.
// GEMM2: gate = sigmoid(H[32,256] @ W2^T + b2) -> global.
// Per ISA 7.12.2 (32-bit operands, wave32):
//   A 16x4 : lanes 0-15 -> M=lane, v0=K0,v1=K1 ; lanes 16-31 -> v0=K2,v1=K3
//   B 4x16 : lanes 0-15 -> N=lane, v0=K0,v1=K1 ; lanes 16-31 -> v0=K2,v1=K3
//   C 16x16: vgpr j -> M = j + 8*(lane/16), N = lane%16
// ---------------------------------------------------------------------------
__global__ __launch_bounds__(1024) void se_mlp(const float* __restrict__ S,
                                               const float* __restrict__ W1,
                                               const float* __restrict__ b1,
                                               const float* __restrict__ W2,
                                               const float* __restrict__ b2,
                                               float* __restrict__ gate) {
    __shared__ float Hs[B * R];            // 32 KB of the 320 KB WGP LDS
    const int tid  = threadIdx.x;
    const int w    = tid >> 5;             // wave id 0..31
    const int lane = tid & 31;
    const int half = lane >> 4;            // 0: K+{0,1}, 1: K+{2,3}
    const int l15  = lane & 15;

    // ---- GEMM1: 2x16 = 32 output tiles of 16x16, one per wave, K=512 ----
    {
        const int mBase = (w >> 4) * 16;
        const int nBase = (w & 15) * 16;
        const float* Arow = S  + (size_t)(mBase + l15) * C;
        const float* Brow = W1 + (size_t)(nBase + l15) * C;  // B[k][n] = W1[n][k]
        v8f acc = {};
        for (int k = 0; k < C; k += 4) {
            const int ka = k + 2 * half;
            v2f a, b;
            a.x = Arow[ka]; a.y = Arow[ka + 1];
            b.x = Brow[ka]; b.y = Brow[ka + 1];
            acc = __builtin_amdgcn_wmma_f32_16x16x4_f32(false, a, false, b,
                                                        (short)0, acc, false, false);
        }
        const int   n    = nBase + l15;
        const float bias = b1[n];
#pragma unroll
        for (int j = 0; j < 8; ++j) {
            const int m = mBase + j + 8 * half;
            const float v = acc[j] + bias;
            Hs[m * R + n] = v > 0.f ? v : 0.f;
        }
    }
    __syncthreads();

    // ---- GEMM2: 2x32 = 64 output tiles, two per wave, K=256 ----
    for (int t = w; t < 64; t += 32) {
        const int mBase = (t & 1) * 16;
        const int nBase = (t >> 1) * 16;
        const float* Arow = Hs + (size_t)(mBase + l15) * R;
        const float* Brow = W2 + (size_t)(nBase + l15) * R;  // B[k][n] = W2[n][k]
        v8f acc = {};
        for (int k = 0; k < R; k += 4) {
            const int ka = k + 2 * half;
            v2f a, b;
            a.x = Arow[ka]; a.y = Arow[ka + 1];
            b.x = Brow[ka]; b.y = Brow[ka + 1];
            acc = __builtin_amdgcn_wmma_f32_16x16x4_f32(false, a, false, b,
                                                        (short)0, acc, false, false);
        }
        const int   n    = nBase + l15;
        const float bias = b2[n];
#pragma unroll
        for (int j = 0; j < 8; ++j) {
            const int m = mBase + j + 8 * half;
            const float v = acc[j] + bias;
            gate[m * C + n] = 1.0f / (1.0f + __expf(-v));
        }
    }
}

// ---------------------------------------------------------------------------
// Kernel 3: scale. One block per (b,c) row, float4 streams.
// Rows walked in REVERSE so the tail of x (still resident in the 192 MB L2
// after the squeeze pass streamed all 256 MB) is re-read from L2, not HBM.
// ---------------------------------------------------------------------------
__global__ __launch_bounds__(256) void se_scale(const float* __restrict__ x,
                                                const float* __restrict__ gate,
                                                float* __restrict__ out) {
    const int row = (int)gridDim.x - 1 - (int)blockIdx.x;  // reverse order
    const float g = gate[row];
    const float4* xr = (const float4*)(x   + (size_t)row * L);
    float4*       orow = (float4*)(out + (size_t)row * L);
    const int tid = threadIdx.x;
#pragma unroll
    for (int i = 0; i < 4; ++i) {
        float4 v = xr[i * 256 + tid];
        v.x *= g; v.y *= g; v.z *= g; v.w *= g;
        orow[i * 256 + tid] = v;
    }
}

// ---------------------------------------------------------------------------
extern "C" void kernel_launch(void* const* d_in, const int* in_sizes, int n_in,
                              void* d_out, int out_size, void* d_ws, size_t ws_size,
                              hipStream_t stream) {
    (void)in_sizes; (void)n_in; (void)out_size; (void)ws_size;
    const float* x  = (const float*)d_in[0];   // [32, 512, 4096]
    const float* W1 = (const float*)d_in[1];   // [256, 512]
    const float* b1 = (const float*)d_in[2];   // [256]
    const float* W2 = (const float*)d_in[3];   // [512, 256]
    const float* b2 = (const float*)d_in[4];   // [512]
    float* out = (float*)d_out;                // [32, 512, 4096]

    float* S    = (float*)d_ws;                // [32*512]  squeeze
    float* gate = S + (size_t)B * C;           // [32*512]  sigmoid gate

    se_squeeze<<<B * C, 256, 0, stream>>>(x, S);
    se_mlp<<<1, 1024, 0, stream>>>(S, W1, b1, W2, b2, gate);
    se_scale<<<B * C, 256, 0, stream>>>(x, gate, out);
}